// GatherModel_42597485642521
// MI455X (gfx1250) — compile-verified
//
#include <hip/hip_runtime.h>
#include <math.h>

#define N_NODES 50000
#define N_EDGES 100000
#define DIN 16
#define KSTEPS 16
#define H 32
#define ED 1024   // H*H

typedef __attribute__((ext_vector_type(2))) float v2f;
typedef __attribute__((ext_vector_type(8))) float v8f;

// ---------------- utility ----------------
__global__ void k_zero(float* __restrict__ p, int n) {
    int i = blockIdx.x * blockDim.x + threadIdx.x;
    if (i < n) p[i] = 0.f;
}

// in-degree via float atomics
__global__ void k_deg(const int* __restrict__ dst, float* __restrict__ deg) {
    int e = blockIdx.x * blockDim.x + threadIdx.x;
    if (e < N_EDGES) atomicAdd(&deg[dst[e]], 1.0f);
}

// degc = (deg==0)?1:deg ; xa = degc ; nf[:,0:16]=n_feat ; nf[:,16]=rwse col0
__global__ void k_degc_init(float* __restrict__ deg, const float* __restrict__ n_feat,
                            float* __restrict__ nf, float* __restrict__ xa) {
    int i = blockIdx.x * blockDim.x + threadIdx.x;
    if (i >= N_NODES) return;
    float d = deg[i];
    d = (d == 0.f) ? 1.f : d;
    deg[i] = d;
    xa[i] = d;
    float* row = nf + (size_t)i * H;
    #pragma unroll
    for (int j = 0; j < DIN; ++j) row[j] = n_feat[(size_t)i * DIN + j];
    row[DIN] = d;
}

__global__ void k_scatter1(const int* __restrict__ src, const int* __restrict__ dst,
                           const float* __restrict__ x, float* __restrict__ y) {
    int e = blockIdx.x * blockDim.x + threadIdx.x;
    if (e < N_EDGES) atomicAdd(&y[dst[e]], x[src[e]]);
}

__global__ void k_rwse_norm(const float* __restrict__ y, const float* __restrict__ degc,
                            float* __restrict__ nf, float* __restrict__ xa, int k) {
    int i = blockIdx.x * blockDim.x + threadIdx.x;
    if (i >= N_NODES) return;
    float v = y[i] / degc[i];
    nf[(size_t)i * H + DIN + k] = v;
    xa[i] = v;
}

// out = relu(nf @ lin0_w + b)  -- one wave per node, lane = out channel
__global__ void k_lin0(const float* __restrict__ nf, const float* __restrict__ w,
                       const float* __restrict__ b, float* __restrict__ out) {
    int lane = threadIdx.x & 31;
    int node = (blockIdx.x * blockDim.x + threadIdx.x) >> 5;
    if (node >= N_NODES) return;
    float x = nf[(size_t)node * H + lane];
    float acc = b[lane];
    #pragma unroll
    for (int h = 0; h < H; ++h)
        acc = fmaf(__shfl(x, h, 32), w[h * H + lane], acc);
    out[(size_t)node * H + lane] = fmaxf(acc, 0.f);
}

// gate[e] = sigmoid(e_feat[e] . gate_w + gate_b)  -- one wave per edge
__global__ void k_gate(const float* __restrict__ e_feat, const float* __restrict__ gw,
                       const float* __restrict__ gb, float* __restrict__ gate) {
    int lane = threadIdx.x & 31;
    int e = (blockIdx.x * blockDim.x + threadIdx.x) >> 5;
    if (e >= N_EDGES) return;
    const float* row = e_feat + (size_t)e * ED;
    float s = 0.f;
    #pragma unroll 4
    for (int j = lane; j < ED; j += 32) s = fmaf(row[j], gw[j], s);
    #pragma unroll
    for (int off = 16; off > 0; off >>= 1) s += __shfl_xor(s, off, 32);
    if (lane == 0) gate[e] = 1.f / (1.f + __expf(-(s + gb[0])));
}

// 1024x1024 f32 transpose (LDS tiled) so B fragments become contiguous b64 loads
__global__ void __launch_bounds__(256)
k_transpose(const float* __restrict__ B, float* __restrict__ Bt) {
    __shared__ float tile[32][33];
    int bx = blockIdx.x * 32, by = blockIdx.y * 32;
    int tx = threadIdx.x, ty = threadIdx.y;   // 32 x 8
    #pragma unroll
    for (int i = 0; i < 32; i += 8)
        tile[ty + i][tx] = B[(size_t)(by + ty + i) * ED + bx + tx];
    __syncthreads();
    #pragma unroll
    for (int i = 0; i < 32; i += 8)
        Bt[(size_t)(bx + ty + i) * ED + by + tx] = tile[tx][ty + i];
}

// W = relu(e_feat @ edge_proj_w + b) * gate  via V_WMMA_F32_16X16X4_F32
// one wave computes a 16(row)x64(col) tile; K loop step 4.
// Bt is edge_proj_w transposed: Bt[n*ED + k], so each B frag is one b64 load.
__global__ void __launch_bounds__(256)
k_edgeproj(const float* __restrict__ A, const float* __restrict__ Bt,
           const float* __restrict__ bias, const float* __restrict__ gate,
           float* __restrict__ Wout) {
    int lane = threadIdx.x & 31;
    int wave = (blockIdx.x * blockDim.x + threadIdx.x) >> 5;
    const int colTiles = ED / 64; // 16
    int rowTile = wave / colTiles;
    int colTile = wave % colTiles;
    if (rowTile >= N_EDGES / 16) return;
    int row0 = rowTile * 16;
    int col0 = colTile * 64;
    int half = lane >> 4;   // 0: K pair {0,1}, 1: K pair {2,3}
    int lm   = lane & 15;

    // A frag: lane(lm) = M, VGPR0/1 = K = 2*half / 2*half+1   (float2 load)
    const float* arow = A  + (size_t)(row0 + lm) * ED + 2 * half;
    // B frag: lane(lm) = N, VGPR0/1 = K = 2*half / 2*half+1   (float2 load from Bt)
    const float* brow = Bt + (size_t)(col0 + lm) * ED + 2 * half;

    v8f c0 = {}, c1 = {}, c2 = {}, c3 = {};
    #pragma unroll 4
    for (int k = 0; k < ED; k += 4) {
        v2f a  = *(const v2f*)(arow + k);
        v2f b0 = *(const v2f*)(brow + k);
        v2f b1 = *(const v2f*)(brow + (size_t)16 * ED + k);
        v2f b2 = *(const v2f*)(brow + (size_t)32 * ED + k);
        v2f b3 = *(const v2f*)(brow + (size_t)48 * ED + k);
        c0 = __builtin_amdgcn_wmma_f32_16x16x4_f32(false, a, false, b0, (short)0, c0, false, false);
        c1 = __builtin_amdgcn_wmma_f32_16x16x4_f32(false, a, false, b1, (short)0, c1, false, false);
        c2 = __builtin_amdgcn_wmma_f32_16x16x4_f32(false, a, false, b2, (short)0, c2, false, false);
        c3 = __builtin_amdgcn_wmma_f32_16x16x4_f32(false, a, false, b3, (short)0, c3, false, false);
    }
    // C/D layout: VGPR r -> M = r + 8*half, N = lm
    #pragma unroll
    for (int r = 0; r < 8; ++r) {
        int row = row0 + r + 8 * half;
        float g = gate[row];
        size_t base = (size_t)row * ED + col0 + lm;
        Wout[base]      = fmaxf(c0[r] + bias[col0 + lm],      0.f) * g;
        Wout[base + 16] = fmaxf(c1[r] + bias[col0 + 16 + lm], 0.f) * g;
        Wout[base + 32] = fmaxf(c2[r] + bias[col0 + 32 + lm], 0.f) * g;
        Wout[base + 48] = fmaxf(c3[r] + bias[col0 + 48 + lm], 0.f) * g;
    }
}

// msg = out[src] @ W[e]; atomic scatter-add into neigh[dst]. one wave per edge.
__global__ void k_msg(const int* __restrict__ src, const int* __restrict__ dst,
                      const float* __restrict__ out, const float* __restrict__ W,
                      float* __restrict__ neigh) {
    int lane = threadIdx.x & 31;
    int e = (blockIdx.x * blockDim.x + threadIdx.x) >> 5;
    if (e >= N_EDGES) return;
    int s = src[e], d = dst[e];
    float x = out[(size_t)s * H + lane];
    const float* We = W + (size_t)e * ED;
    float acc = 0.f;
    #pragma unroll
    for (int h = 0; h < H; ++h)
        acc = fmaf(__shfl(x, h, 32), We[h * H + lane], acc);
    atomicAdd(&neigh[(size_t)d * H + lane], acc);
}

// m = relu(neigh/degc + out + cb); out_new = [m,out] @ mw + mb
__global__ void k_node_update(const float* __restrict__ neigh, const float* __restrict__ degc,
                              const float* __restrict__ outp, const float* __restrict__ cb,
                              const float* __restrict__ mw, const float* __restrict__ mb,
                              float* __restrict__ outn) {
    int lane = threadIdx.x & 31;
    int n = (blockIdx.x * blockDim.x + threadIdx.x) >> 5;
    if (n >= N_NODES) return;
    float o = outp[(size_t)n * H + lane];
    float m = fmaxf(neigh[(size_t)n * H + lane] / degc[n] + o + cb[lane], 0.f);
    float acc = mb[lane];
    #pragma unroll
    for (int h = 0; h < H; ++h)
        acc = fmaf(__shfl(m, h, 32), mw[h * H + lane], acc);
    #pragma unroll
    for (int h = 0; h < H; ++h)
        acc = fmaf(__shfl(o, h, 32), mw[(H + h) * H + lane], acc);
    outn[(size_t)n * H + lane] = acc;
}

// group pooling scatter
__global__ void k_group(const int* __restrict__ src, const int* __restrict__ dst,
                        const float* __restrict__ out, float* __restrict__ grp) {
    int lane = threadIdx.x & 31;
    int e = (blockIdx.x * blockDim.x + threadIdx.x) >> 5;
    if (e >= N_EDGES) return;
    atomicAdd(&grp[(size_t)dst[e] * H + lane], out[(size_t)src[e] * H + lane]);
}

// res = [out, grp/degc] @ sw + sb + init(nf)
__global__ void k_final(const float* __restrict__ out, const float* __restrict__ grp,
                        const float* __restrict__ degc, const float* __restrict__ sw,
                        const float* __restrict__ sb, const float* __restrict__ nf,
                        float* __restrict__ res) {
    int lane = threadIdx.x & 31;
    int n = (blockIdx.x * blockDim.x + threadIdx.x) >> 5;
    if (n >= N_NODES) return;
    float o = out[(size_t)n * H + lane];
    float g = grp[(size_t)n * H + lane] / degc[n];
    float acc = sb[lane];
    #pragma unroll
    for (int h = 0; h < H; ++h)
        acc = fmaf(__shfl(o, h, 32), sw[h * H + lane], acc);
    #pragma unroll
    for (int h = 0; h < H; ++h)
        acc = fmaf(__shfl(g, h, 32), sw[(H + h) * H + lane], acc);
    res[(size_t)n * H + lane] = acc + nf[(size_t)n * H + lane];
}

extern "C" void kernel_launch(void* const* d_in, const int* in_sizes, int n_in,
                              void* d_out, int out_size, void* d_ws, size_t ws_size,
                              hipStream_t stream) {
    const float* n_feat      = (const float*)d_in[0];
    const float* e_feat      = (const float*)d_in[1];
    const int*   src         = (const int*)d_in[2];
    const int*   dst         = (const int*)d_in[3];
    const float* edge_proj_w = (const float*)d_in[4];
    const float* edge_proj_b = (const float*)d_in[5];
    const float* gate_w      = (const float*)d_in[6];
    const float* gate_b      = (const float*)d_in[7];
    const float* lin0_w      = (const float*)d_in[8];
    const float* lin0_b      = (const float*)d_in[9];
    const float* conv_bias   = (const float*)d_in[10];
    const float* message_w   = (const float*)d_in[11];
    const float* message_b   = (const float*)d_in[12];
    const float* subgraph_w  = (const float*)d_in[13];
    const float* subgraph_b  = (const float*)d_in[14];

    char* p = (char*)d_ws;
    auto carve = [&](size_t bytes) -> void* {
        void* r = (void*)p;
        p += (bytes + 255) & ~(size_t)255;
        return r;
    };
    float* Wbuf  = (float*)carve((size_t)N_EDGES * ED * sizeof(float)); // 409.6 MB
    float* Bt    = (float*)carve((size_t)ED * ED * sizeof(float));      // 4 MB
    float* gate  = (float*)carve((size_t)N_EDGES * sizeof(float));
    float* degc  = (float*)carve((size_t)N_NODES * sizeof(float));
    float* xa    = (float*)carve((size_t)N_NODES * sizeof(float));
    float* xb    = (float*)carve((size_t)N_NODES * sizeof(float));
    float* nf    = (float*)carve((size_t)N_NODES * H * sizeof(float)); // init
    float* outA  = (float*)carve((size_t)N_NODES * H * sizeof(float));
    float* outB  = (float*)carve((size_t)N_NODES * H * sizeof(float));
    float* neigh = (float*)carve((size_t)N_NODES * H * sizeof(float));

    dim3 b256(256);
    auto gN = [](long n) { return dim3((unsigned)((n + 255) / 256)); };

    // degrees
    k_zero<<<gN(N_NODES), b256, 0, stream>>>(degc, N_NODES);
    k_deg<<<gN(N_EDGES), b256, 0, stream>>>(dst, degc);
    k_degc_init<<<gN(N_NODES), b256, 0, stream>>>(degc, n_feat, nf, xa);

    // RWSE random walk columns 1..15
    for (int k = 1; k < KSTEPS; ++k) {
        k_zero<<<gN(N_NODES), b256, 0, stream>>>(xb, N_NODES);
        k_scatter1<<<gN(N_EDGES), b256, 0, stream>>>(src, dst, xa, xb);
        k_rwse_norm<<<gN(N_NODES), b256, 0, stream>>>(xb, degc, nf, xa, k);
    }

    // node encoder
    k_lin0<<<gN((long)N_NODES * 32), b256, 0, stream>>>(nf, lin0_w, lin0_b, outA);

    // edge gate + weight transpose + edge-conditioned weights (the big WMMA GEMM)
    k_gate<<<gN((long)N_EDGES * 32), b256, 0, stream>>>(e_feat, gate_w, gate_b, gate);
    k_transpose<<<dim3(ED / 32, ED / 32), dim3(32, 8), 0, stream>>>(edge_proj_w, Bt);
    k_edgeproj<<<dim3((N_EDGES / 16) * (ED / 64) / 8), b256, 0, stream>>>(
        e_feat, Bt, edge_proj_b, gate, Wbuf);

    // 3 rounds of NNConv mean-aggregation message passing
    float* oA = outA; float* oB = outB;
    for (int it = 0; it < 3; ++it) {
        k_zero<<<gN((long)N_NODES * H), b256, 0, stream>>>(neigh, N_NODES * H);
        k_msg<<<gN((long)N_EDGES * 32), b256, 0, stream>>>(src, dst, oA, Wbuf, neigh);
        k_node_update<<<gN((long)N_NODES * 32), b256, 0, stream>>>(
            neigh, degc, oA, conv_bias, message_w, message_b, oB);
        float* t = oA; oA = oB; oB = t;
    }

    // group pooling + subgraph projection + residual
    k_zero<<<gN((long)N_NODES * H), b256, 0, stream>>>(neigh, N_NODES * H);
    k_group<<<gN((long)N_EDGES * 32), b256, 0, stream>>>(src, dst, oA, neigh);
    k_final<<<gN((long)N_NODES * 32), b256, 0, stream>>>(
        oA, neigh, degc, subgraph_w, subgraph_b, nf, (float*)d_out);
}